// Model_69080253989304
// MI455X (gfx1250) — compile-verified
//
#include <hip/hip_runtime.h>

#ifndef __has_builtin
#define __has_builtin(x) 0
#endif

#define SEQ   4096
#define BATCH 8192
#define HID   4
#define TPB   32     // one wave32 per workgroup
#define DEPTH 32     // async ring depth (<= 63, ASYNCcnt is 6-bit)

#if __has_builtin(__builtin_amdgcn_global_load_async_to_lds_b32) && \
    __has_builtin(__builtin_amdgcn_s_wait_asynccnt)
#define HAVE_ASYNC 1
#else
#define HAVE_ASYNC 0
#warning "gfx1250 async-to-LDS builtins not found; using synchronous staging fallback"
#endif

#if __has_builtin(__builtin_amdgcn_tanhf)
#define HAVE_TANH 1
#else
#define HAVE_TANH 0
#warning "native v_tanh_f32 builtin not found; using exp2/rcp tanh chain"
#endif

typedef __attribute__((address_space(1))) int g_i32;  // global
typedef __attribute__((address_space(3))) int l_i32;  // LDS

// Issue one 4-byte async global->LDS copy for this lane (ASYNCcnt-tracked).
__device__ __forceinline__ void stage_one(const float* gsrc, float* ldst) {
#if HAVE_ASYNC
  __builtin_amdgcn_global_load_async_to_lds_b32(
      (g_i32*)(__attribute__((address_space(1))) void*)(void*)const_cast<float*>(gsrc),
      (l_i32*)(__attribute__((address_space(3))) void*)(void*)ldst,
      /*offset=*/0, /*cpol=*/0);
#else
  *ldst = __builtin_nontemporal_load(gsrc);
#endif
}

__device__ __forceinline__ void wait_ring_oldest_done() {
#if HAVE_ASYNC
  __builtin_amdgcn_s_wait_asynccnt(DEPTH - 1);  // oldest outstanding is complete
#endif
  asm volatile("" ::: "memory");
}

__device__ __forceinline__ void wait_all_async() {
#if HAVE_ASYNC
  __builtin_amdgcn_s_wait_asynccnt(0);
#endif
  asm volatile("" ::: "memory");
}

// Activation. Native path: single v_tanh_f32 trans op (shortest serial chain).
// Fallback path expects z pre-scaled by 2*log2(e): tanh = 1 - 2/(exp2(z')+1).
__device__ __forceinline__ float activation(float z) {
#if HAVE_TANH
  return __builtin_amdgcn_tanhf(z);
#else
  const float e = __builtin_amdgcn_exp2f(z);
  const float r = __builtin_amdgcn_rcpf(e + 1.0f);
  return __builtin_fmaf(-2.0f, r, 1.0f);
#endif
}

__global__ __launch_bounds__(TPB) void rnn4_scan_kernel(
    const float* __restrict__ x,     // (SEQ, BATCH, 1)
    const float* __restrict__ h0,    // (1, BATCH, HID)
    const float* __restrict__ W_ih,  // (HID, 1)
    const float* __restrict__ b_ih,  // (HID,)
    const float* __restrict__ W_hh,  // (HID, HID) row-major
    const float* __restrict__ b_hh,  // (HID,)
    const float* __restrict__ fc_W,  // (1, HID)
    const float* __restrict__ fc_b,  // (1,)
    float* __restrict__ out)         // [0,BATCH) = y_last ; [BATCH,...) = h_T
{
  __shared__ float ring[DEPTH * TPB];

  const int lane = threadIdx.x;            // 0..31
  const int b    = blockIdx.x * TPB + lane;

#if HAVE_TANH
  const float C = 1.0f;                    // native tanh takes plain z
#else
  const float C = 2.885390081777926815f;   // 2*log2(e), folded into weights
#endif

  float wih[HID], bias[HID], whh[HID][HID];
#pragma unroll
  for (int i = 0; i < HID; ++i) {
    wih[i]  = C * W_ih[i];
    bias[i] = C * (b_ih[i] + b_hh[i]);   // xp bias + recurrent bias
#pragma unroll
    for (int j = 0; j < HID; ++j) whh[i][j] = C * W_hh[i * HID + j];
  }

  float h[HID];
#pragma unroll
  for (int i = 0; i < HID; ++i) h[i] = h0[b * HID + i];

  float* slot = &ring[lane];               // this lane's column in the ring
  const float* gp = x + b;                 // x[s*BATCH + b]

  // ---- prologue: fill the async ring DEPTH steps deep ----
#pragma unroll
  for (int k = 0; k < DEPTH; ++k)
    stage_one(gp + (size_t)k * BATCH, slot + (k & (DEPTH - 1)) * TPB);

  int s = 0;
  // ---- steady state: wait oldest, consume, re-issue DEPTH ahead ----
  for (; s < SEQ - DEPTH; ++s) {
    wait_ring_oldest_done();
    const float xv = slot[(s & (DEPTH - 1)) * TPB];
    asm volatile("" ::: "memory");         // read slot before re-issuing into it
    stage_one(gp + (size_t)(s + DEPTH) * BATCH, slot + (s & (DEPTH - 1)) * TPB);

    float z[HID];
#pragma unroll
    for (int i = 0; i < HID; ++i) {
      float zi = __builtin_fmaf(xv, wih[i], bias[i]);
      zi = __builtin_fmaf(h[0], whh[i][0], zi);
      zi = __builtin_fmaf(h[1], whh[i][1], zi);
      zi = __builtin_fmaf(h[2], whh[i][2], zi);
      zi = __builtin_fmaf(h[3], whh[i][3], zi);
      z[i] = zi;
    }
#pragma unroll
    for (int i = 0; i < HID; ++i) h[i] = activation(z[i]);
  }

  // ---- tail: everything already in flight; drain once, then consume ----
  wait_all_async();
  for (; s < SEQ; ++s) {
    const float xv = slot[(s & (DEPTH - 1)) * TPB];
    float z[HID];
#pragma unroll
    for (int i = 0; i < HID; ++i) {
      float zi = __builtin_fmaf(xv, wih[i], bias[i]);
      zi = __builtin_fmaf(h[0], whh[i][0], zi);
      zi = __builtin_fmaf(h[1], whh[i][1], zi);
      zi = __builtin_fmaf(h[2], whh[i][2], zi);
      zi = __builtin_fmaf(h[3], whh[i][3], zi);
      z[i] = zi;
    }
#pragma unroll
    for (int i = 0; i < HID; ++i) h[i] = activation(z[i]);
  }

  // ---- epilogue: y = fc(h_T), and emit h_T ----
  float y = fc_b[0];
#pragma unroll
  for (int i = 0; i < HID; ++i) y = __builtin_fmaf(h[i], fc_W[i], y);

  out[b] = y;                                    // y_last (BATCH,1)
#pragma unroll
  for (int i = 0; i < HID; ++i)
    out[BATCH + b * HID + i] = h[i];             // hn (1,BATCH,HID)
}

extern "C" void kernel_launch(void* const* d_in, const int* in_sizes, int n_in,
                              void* d_out, int out_size, void* d_ws, size_t ws_size,
                              hipStream_t stream) {
  const float* x    = (const float*)d_in[0];
  const float* h0   = (const float*)d_in[1];
  const float* W_ih = (const float*)d_in[2];
  const float* b_ih = (const float*)d_in[3];
  const float* W_hh = (const float*)d_in[4];
  const float* b_hh = (const float*)d_in[5];
  const float* fc_W = (const float*)d_in[6];
  const float* fc_b = (const float*)d_in[7];
  float* out = (float*)d_out;

  dim3 grid(BATCH / TPB);   // 256 single-wave workgroups -> spread across WGPs
  dim3 block(TPB);
  rnn4_scan_kernel<<<grid, block, 0, stream>>>(x, h0, W_ih, b_ih, W_hh, b_hh,
                                               fc_W, fc_b, out);
}